// GATSingleLayer_9818295239349
// MI455X (gfx1250) — compile-verified
//
#include <hip/hip_runtime.h>
#include <math.h>

typedef __attribute__((ext_vector_type(2))) float v2f;
typedef __attribute__((ext_vector_type(8))) float v8f;

#define HEADS 8
#define CHEAD 16
#define FDIM 128
#define NEG_SLOPE 0.2f

// ---------------------------------------------------------------------------
// Kernel A: xp[N,128] = x[N,128] @ W[128,128] using V_WMMA_F32_16X16X4_F32.
// Block = 256 threads (8 waves). blockIdx.x selects a 16-row (node) tile,
// wave w owns the 16-column tile n0 = w*16 (i.e., head w), K looped 128/4.
// Fragment layouts per CDNA5 ISA 7.12.2 (32-bit A 16x4 / B 4x16 / D 16x16).
// ---------------------------------------------------------------------------
__global__ __launch_bounds__(256) void k_gemm(const float* __restrict__ x,
                                              const float* __restrict__ W,
                                              float* __restrict__ xp) {
  const int wave = threadIdx.x >> 5;
  const int lane = threadIdx.x & 31;
  const int half = lane >> 4;    // 0: lanes 0-15, 1: lanes 16-31
  const int l    = lane & 15;
  const int m0 = blockIdx.x * 16;
  const int n0 = wave * 16;

  const float* xrow = x + (size_t)(m0 + l) * FDIM;
  v8f acc = {0.f, 0.f, 0.f, 0.f, 0.f, 0.f, 0.f, 0.f};

#pragma unroll
  for (int k0 = 0; k0 < FDIM; k0 += 4) {
    const int ka = k0 + 2 * half;
    v2f a, b;
    // A (16x4 f32): lane group half: VGPR0=K(ka), VGPR1=K(ka+1), row = l
    a.x = xrow[ka];
    a.y = xrow[ka + 1];
    // B (4x16 f32): VGPR0 = row K(ka), VGPR1 = row K(ka+1), col = l
    b.x = W[(size_t)ka * FDIM + n0 + l];
    b.y = W[(size_t)(ka + 1) * FDIM + n0 + l];
    acc = __builtin_amdgcn_wmma_f32_16x16x4_f32(
        /*neg_a=*/false, a, /*neg_b=*/false, b,
        /*c_mod=*/(short)0, acc, /*reuse_a=*/false, /*reuse_b=*/false);
  }

  // D (16x16 f32): VGPR r -> row (r + 8*half), col = l
#pragma unroll
  for (int r = 0; r < 8; ++r) {
    xp[(size_t)(m0 + r + 8 * half) * FDIM + n0 + l] = acc[r];
  }
}

// Order-preserving float <-> uint mapping (for atomic segment-max)
__device__ __forceinline__ unsigned f2ord(float f) {
  unsigned u = __float_as_uint(f);
  return (u & 0x80000000u) ? ~u : (u | 0x80000000u);
}
__device__ __forceinline__ float ord2f(unsigned u) {
  unsigned b = (u & 0x80000000u) ? (u & 0x7FFFFFFFu) : ~u;
  return __uint_as_float(b);
}

#define ORD_NEG_INF 0x007FFFFFu  // f2ord(-inf)

// ---------------------------------------------------------------------------
// Kernel B: per (node, head): a_src/a_dst logits; init m=-inf, denom=0,
// out = bias.
// ---------------------------------------------------------------------------
__global__ __launch_bounds__(256) void k_node(const float* __restrict__ xp,
                                              const float* __restrict__ att_src,
                                              const float* __restrict__ att_dst,
                                              const float* __restrict__ bias,
                                              float* __restrict__ a_src,
                                              float* __restrict__ a_dst,
                                              unsigned* __restrict__ mord,
                                              float* __restrict__ denom,
                                              float* __restrict__ out, int NH) {
  int t = blockIdx.x * blockDim.x + threadIdx.x;
  if (t >= NH) return;
  int n = t >> 3, h = t & 7;
  const float* row = xp + (size_t)n * FDIM + h * CHEAD;
  float s = 0.f, d = 0.f;
#pragma unroll
  for (int c = 0; c < CHEAD; ++c) {
    float v = row[c];
    s += v * att_src[h * CHEAD + c];
    d += v * att_dst[h * CHEAD + c];
  }
  a_src[t] = s;
  a_dst[t] = d;
  mord[t] = ORD_NEG_INF;
  denom[t] = 0.f;
  float* o = out + (size_t)n * FDIM + h * CHEAD;
#pragma unroll
  for (int c = 0; c < CHEAD; ++c) o[c] = bias[h * CHEAD + c];
}

__device__ __forceinline__ float lrelu(float e) {
  return (e > 0.f) ? e : NEG_SLOPE * e;
}

// ---------------------------------------------------------------------------
// Kernel C: per (edge, head) segment max via ordered-uint atomicMax.
// ---------------------------------------------------------------------------
__global__ __launch_bounds__(256) void k_edge_max(const int* __restrict__ ei,
                                                  const float* __restrict__ a_src,
                                                  const float* __restrict__ a_dst,
                                                  unsigned* __restrict__ mord,
                                                  int E) {
  int t = blockIdx.x * blockDim.x + threadIdx.x;
  int edge = t >> 3;
  if (edge >= E) return;
  int h = t & 7;
  int src = ei[edge], dst = ei[E + edge];
  float e = lrelu(a_src[src * HEADS + h] + a_dst[dst * HEADS + h]);
  atomicMax(&mord[dst * HEADS + h], f2ord(e));
}

// ---------------------------------------------------------------------------
// Kernel C2: decode ordered max -> float; empty segments (-inf) -> 0.
// In-place reinterpret of the same buffer.
// ---------------------------------------------------------------------------
__global__ __launch_bounds__(256) void k_finalize_max(unsigned* __restrict__ mord,
                                                      float* __restrict__ m, int NH) {
  int t = blockIdx.x * blockDim.x + threadIdx.x;
  if (t >= NH) return;
  float v = ord2f(mord[t]);
  // non-finite (exp == 0xFF) -> 0
  if ((__float_as_uint(v) & 0x7F800000u) == 0x7F800000u) v = 0.f;
  m[t] = v;
}

// ---------------------------------------------------------------------------
// Kernel D: per (edge, head) denom += exp(e - m[dst]).
// ---------------------------------------------------------------------------
__global__ __launch_bounds__(256) void k_edge_expsum(const int* __restrict__ ei,
                                                     const float* __restrict__ a_src,
                                                     const float* __restrict__ a_dst,
                                                     const float* __restrict__ m,
                                                     float* __restrict__ denom,
                                                     int E) {
  int t = blockIdx.x * blockDim.x + threadIdx.x;
  int edge = t >> 3;
  if (edge >= E) return;
  int h = t & 7;
  int src = ei[edge], dst = ei[E + edge];
  float e = lrelu(a_src[src * HEADS + h] + a_dst[dst * HEADS + h]);
  float ex = __expf(e - m[dst * HEADS + h]);
  atomicAdd(&denom[dst * HEADS + h], ex);
}

// ---------------------------------------------------------------------------
// Kernel E: per (edge, feature) weighted scatter-aggregate.
// 128 threads per edge; alpha recomputed per lane (uniform per 16-lane group,
// broadcast loads). out is L2-resident on MI455X (51 MB << 192 MB L2), so the
// float atomic adds resolve in L2.
// ---------------------------------------------------------------------------
__global__ __launch_bounds__(256) void k_aggregate(const int* __restrict__ ei,
                                                   const float* __restrict__ a_src,
                                                   const float* __restrict__ a_dst,
                                                   const float* __restrict__ m,
                                                   const float* __restrict__ denom,
                                                   const float* __restrict__ xp,
                                                   float* __restrict__ out, int E) {
  int t = blockIdx.x * blockDim.x + threadIdx.x;
  int edge = t >> 7;
  if (edge >= E) return;
  int r = t & 127;        // feature index 0..127
  int h = r >> 4;         // head
  int src = ei[edge], dst = ei[E + edge];
  float e = lrelu(a_src[src * HEADS + h] + a_dst[dst * HEADS + h]);
  float ex = __expf(e - m[dst * HEADS + h]);
  float alpha = ex / (denom[dst * HEADS + h] + 1e-16f);
  atomicAdd(&out[(size_t)dst * FDIM + r], xp[(size_t)src * FDIM + r] * alpha);
}

// ---------------------------------------------------------------------------
extern "C" void kernel_launch(void* const* d_in, const int* in_sizes, int n_in,
                              void* d_out, int out_size, void* d_ws, size_t ws_size,
                              hipStream_t stream) {
  const float* x       = (const float*)d_in[0];
  const int*   ei      = (const int*)d_in[1];
  const float* W       = (const float*)d_in[2];
  const float* att_src = (const float*)d_in[3];
  const float* att_dst = (const float*)d_in[4];
  const float* bias    = (const float*)d_in[5];
  float*       out     = (float*)d_out;

  const int N = in_sizes[0] / FDIM;   // 100000
  const int E = in_sizes[1] / 2;      // 1600000
  const int NH = N * HEADS;

  // Workspace layout (floats): xp | a_src | a_dst | m(ord/float) | denom
  float* xp    = (float*)d_ws;
  float* a_src = xp + (size_t)N * FDIM;
  float* a_dst = a_src + (size_t)NH;
  float* mbuf  = a_dst + (size_t)NH;
  float* denom = mbuf + (size_t)NH;

  // A: projection GEMM (WMMA). N = 100000 is a multiple of 16.
  k_gemm<<<N / 16, 256, 0, stream>>>(x, W, xp);

  // B: node logits + init
  k_node<<<(NH + 255) / 256, 256, 0, stream>>>(xp, att_src, att_dst, bias,
                                               a_src, a_dst, (unsigned*)mbuf,
                                               denom, out, NH);

  // C: segment max over incoming edges
  k_edge_max<<<(E * HEADS + 255) / 256, 256, 0, stream>>>(ei, a_src, a_dst,
                                                          (unsigned*)mbuf, E);

  // C2: decode max, zero empty segments
  k_finalize_max<<<(NH + 255) / 256, 256, 0, stream>>>((unsigned*)mbuf, mbuf, NH);

  // D: softmax denominator
  k_edge_expsum<<<(E * HEADS + 255) / 256, 256, 0, stream>>>(ei, a_src, a_dst,
                                                             mbuf, denom, E);

  // E: weighted scatter-aggregate (E*128 threads)
  int agg_blocks = (int)(((long long)E * FDIM + 255) / 256);
  k_aggregate<<<agg_blocks, 256, 0, stream>>>(ei, a_src, a_dst, mbuf, denom,
                                              xp, out, E);
}